// GATConvPyG_88261577933304
// MI455X (gfx1250) — compile-verified
//
#include <hip/hip_runtime.h>
#include <math.h>

typedef __attribute__((ext_vector_type(2))) float v2f;
typedef __attribute__((ext_vector_type(8))) float v8f;

#define F   128      // F_IN == H*FO
#define H   4
#define FO  32
#define NEG 0.2f
#define LDSS 132     // padded LDS row stride (floats) to avoid 64-bank conflicts

__device__ __forceinline__ float leaky(float v) { return v > 0.f ? v : NEG * v; }

// float atomic max via int/uint bit tricks (emax initialized to -inf)
__device__ __forceinline__ void atomicMaxF(float* addr, float val) {
    if (val >= 0.f) atomicMax((int*)addr, __float_as_int(val));
    else            atomicMin((unsigned int*)addr, (unsigned int)__float_as_int(val));
}

// ---------------------------------------------------------------------------
// Kernel 1: x = feat @ W  (to workspace), out = feat @ W_res (residual, to d_out)
// 256 threads / 8 waves per block; block owns 64 rows of feat staged in LDS.
// Each wave runs 16x16 tiles with V_WMMA_F32_16X16X4_F32 over K=128.
// Full-tile blocks (all but the last) take a branch-free fast path.
// ---------------------------------------------------------------------------
__global__ __launch_bounds__(256)
void gat_proj(const float* __restrict__ feat, const float* __restrict__ W,
              const float* __restrict__ Wres, float* __restrict__ x,
              float* __restrict__ out, int N)
{
    __shared__ float tile[64 * LDSS];
    const int t = threadIdx.x;
    const int blockRow = blockIdx.x * 64;
    const bool fullTile = (blockRow + 64) <= N;   // wave-uniform (scalar branch)

    // cooperative load: 64 rows x 128 cols = 2048 float4 slots / 256 threads
    if (fullTile) {
        #pragma unroll
        for (int i = 0; i < 8; ++i) {
            int idx = t + i * 256;
            int r   = idx >> 5;          // 32 float4 per row
            int c4  = idx & 31;
            float4 v = *(const float4*)(feat + (size_t)(blockRow + r) * F + c4 * 4);
            *(float4*)(&tile[r * LDSS + c4 * 4]) = v;
        }
    } else {
        #pragma unroll
        for (int i = 0; i < 8; ++i) {
            int idx = t + i * 256;
            int r   = idx >> 5;
            int c4  = idx & 31;
            float4 v = make_float4(0.f, 0.f, 0.f, 0.f);
            int gr = blockRow + r;
            if (gr < N) v = *(const float4*)(feat + (size_t)gr * F + c4 * 4);
            *(float4*)(&tile[r * LDSS + c4 * 4]) = v;
        }
    }
    __syncthreads();

    const int lane = t & 31;
    const int wave = t >> 5;
    const int half = lane >> 4;   // 0: lanes 0-15, 1: lanes 16-31
    const int l16  = lane & 15;

    // 64 tile jobs: job[5]=matrix select, job[4:3]=M tile (0..3), job[2:0]=N tile (0..7)
    for (int job = wave; job < 64; job += 8) {
        const int matsel = job >> 5;
        const int tid    = job & 31;
        const int m0 = (tid >> 3) * 16;
        const int n0 = (tid & 7)  * 16;
        const float* __restrict__ B = matsel ? Wres : W;
        float*       __restrict__ D = matsel ? out  : x;

        v8f acc = {};
        #pragma unroll 8
        for (int kk = 0; kk < F; kk += 4) {
            const int ks = kk + 2 * half;
            // A 16x4 frag: lane m = l16, (K=ks, ks+1) — contiguous in LDS row
            v2f a = *(const v2f*)(&tile[(m0 + l16) * LDSS + ks]);
            // B 4x16 frag: row K=ks (vgpr0) / K=ks+1 (vgpr1), col = n0+l16
            v2f b;
            b.x = B[(size_t)ks * F + n0 + l16];
            b.y = B[(size_t)(ks + 1) * F + n0 + l16];
            acc = __builtin_amdgcn_wmma_f32_16x16x4_f32(false, a, false, b,
                                                        (short)0, acc, false, false);
        }

        // D layout: vgpr j -> row m0 + j + 8*half, col n0 + l16
        float* __restrict__ Dp = D + (size_t)(blockRow + m0 + 8 * half) * F + n0 + l16;
        if (fullTile) {
            #pragma unroll
            for (int j = 0; j < 8; ++j)
                Dp[(size_t)j * F] = acc[j];
        } else {
            #pragma unroll
            for (int j = 0; j < 8; ++j) {
                int gr = blockRow + m0 + j + 8 * half;
                if (gr < N) Dp[(size_t)j * F] = acc[j];
            }
        }
    }
}

// ---------------------------------------------------------------------------
// Kernel 2: per-node attention logits a_s/a_d [N,H] + init emax/-inf, denom/0
// One wave per node; coalesced row read; shfl_xor tree reduction.
// ---------------------------------------------------------------------------
__global__ __launch_bounds__(256)
void gat_logits(const float* __restrict__ x, const float* __restrict__ att_s,
                const float* __restrict__ att_d, float* __restrict__ a_s,
                float* __restrict__ a_d, float* __restrict__ emax,
                float* __restrict__ denom, int N)
{
    const int wave = threadIdx.x >> 5, lane = threadIdx.x & 31;
    const int n = blockIdx.x * 8 + wave;
    if (n >= N) return;

    float s[H], d[H];
    #pragma unroll
    for (int h = 0; h < H; ++h) {
        float xv = x[(size_t)n * F + h * FO + lane];
        s[h] = xv * att_s[h * FO + lane];
        d[h] = xv * att_d[h * FO + lane];
    }
    #pragma unroll
    for (int off = 16; off > 0; off >>= 1) {
        #pragma unroll
        for (int h = 0; h < H; ++h) {
            s[h] += __shfl_xor(s[h], off, 32);
            d[h] += __shfl_xor(d[h], off, 32);
        }
    }
    if (lane == 0) {
        #pragma unroll
        for (int h = 0; h < H; ++h) {
            a_s[n * H + h]   = s[h];
            a_d[n * H + h]   = d[h];
            emax[n * H + h]  = -INFINITY;
            denom[n * H + h] = 0.f;
        }
    }
}

// ---------------------------------------------------------------------------
// Kernel 3: segment max over incoming edges (atomic float max, L2-resident)
// ---------------------------------------------------------------------------
__global__ __launch_bounds__(256)
void gat_edge_max(const int* __restrict__ src, const int* __restrict__ dst,
                  const float* __restrict__ a_s, const float* __restrict__ a_d,
                  float* __restrict__ emax, int E)
{
    int e = blockIdx.x * blockDim.x + threadIdx.x;
    if (e >= E) return;
    int s = src[e], d = dst[e];
    #pragma unroll
    for (int h = 0; h < H; ++h)
        atomicMaxF(&emax[d * H + h], leaky(a_s[s * H + h] + a_d[d * H + h]));
}

// ---------------------------------------------------------------------------
// Kernel 4: segment sum of exp(e - emax[dst])  (recompute e; no per-edge store)
// ---------------------------------------------------------------------------
__global__ __launch_bounds__(256)
void gat_edge_sum(const int* __restrict__ src, const int* __restrict__ dst,
                  const float* __restrict__ a_s, const float* __restrict__ a_d,
                  const float* __restrict__ emax, float* __restrict__ denom, int E)
{
    int e = blockIdx.x * blockDim.x + threadIdx.x;
    if (e >= E) return;
    int s = src[e], d = dst[e];
    #pragma unroll
    for (int h = 0; h < H; ++h) {
        float v = leaky(a_s[s * H + h] + a_d[d * H + h]);
        atomicAdd(&denom[d * H + h], __expf(v - emax[d * H + h]));
    }
}

// ---------------------------------------------------------------------------
// Kernel 5: out[dst] += alpha * x[src]  — one wave per edge, lane owns 4 cols
// (all 4 cols of a lane are in the same head: h = lane>>3). d_out already
// holds the residual projection from kernel 1.
// ---------------------------------------------------------------------------
__global__ __launch_bounds__(256)
void gat_aggregate(const int* __restrict__ src, const int* __restrict__ dst,
                   const float* __restrict__ x, const float* __restrict__ a_s,
                   const float* __restrict__ a_d, const float* __restrict__ emax,
                   const float* __restrict__ denom, float* __restrict__ out, int E)
{
    const int wave = threadIdx.x >> 5, lane = threadIdx.x & 31;
    const int e = blockIdx.x * 8 + wave;
    if (e >= E) return;
    const int s = src[e], d = dst[e];
    const int h = lane >> 3;

    float logit = leaky(a_s[s * H + h] + a_d[d * H + h]);
    float alpha = __expf(logit - emax[d * H + h]) / denom[d * H + h];

    float4 xv = *(const float4*)(x + (size_t)s * F + lane * 4);
    float* o = out + (size_t)d * F + lane * 4;
    atomicAdd(o + 0, xv.x * alpha);
    atomicAdd(o + 1, xv.y * alpha);
    atomicAdd(o + 2, xv.z * alpha);
    atomicAdd(o + 3, xv.w * alpha);
}

// ---------------------------------------------------------------------------
extern "C" void kernel_launch(void* const* d_in, const int* in_sizes, int n_in,
                              void* d_out, int out_size, void* d_ws, size_t ws_size,
                              hipStream_t stream)
{
    const float* feat  = (const float*)d_in[0];
    const float* W     = (const float*)d_in[1];
    const float* att_s = (const float*)d_in[2];
    const float* att_d = (const float*)d_in[3];
    const float* Wres  = (const float*)d_in[4];
    const int*   src   = (const int*)d_in[5];
    const int*   dst   = (const int*)d_in[6];

    const int N = in_sizes[0] / F;   // 100000
    const int E = in_sizes[5];       // 1600000
    float* out = (float*)d_out;

    // workspace layout: x[N*128] | a_s[N*4] | a_d[N*4] | emax[N*4] | denom[N*4]
    float* x     = (float*)d_ws;
    float* a_s   = x     + (size_t)N * F;
    float* a_d   = a_s   + (size_t)N * H;
    float* emax  = a_d   + (size_t)N * H;
    float* denom = emax  + (size_t)N * H;

    gat_proj     <<<(N + 63) / 64,   256, 0, stream>>>(feat, W, Wres, x, out, N);
    gat_logits   <<<(N + 7) / 8,     256, 0, stream>>>(x, att_s, att_d, a_s, a_d, emax, denom, N);
    gat_edge_max <<<(E + 255) / 256, 256, 0, stream>>>(src, dst, a_s, a_d, emax, E);
    gat_edge_sum <<<(E + 255) / 256, 256, 0, stream>>>(src, dst, a_s, a_d, emax, denom, E);
    gat_aggregate<<<(E + 7) / 8,     256, 0, stream>>>(src, dst, x, a_s, a_d, emax, denom, out, E);
}